// BallQLoss_12841952215429
// MI455X (gfx1250) — compile-verified
//
#include <hip/hip_runtime.h>

typedef __attribute__((ext_vector_type(2))) float v2f;
typedef __attribute__((ext_vector_type(8))) float v8f;

#define NPTS  4096
#define BATCH 4
#define CH    30
#define KN    16
#define R2    0.04f   // RADIUS^2 = 0.2^2

// One wave (32 threads) per block; each block handles a tile of 16 query points
// of one batch. Distance tiles computed with V_WMMA_F32_16X16X4_F32 (K padded
// 3->4, A pre-scaled by -2 so d2 = |q|^2 + |c|^2 + wmma_result).
__global__ __launch_bounds__(32) void ballq_loss_kernel(const float* __restrict__ pc,
                                                        const float* __restrict__ mask,
                                                        float* __restrict__ out) {
  __shared__ float    sq_q[16];        // |query|^2 per tile row
  __shared__ unsigned balS[8];         // per-r ballot of (d2 < R2)
  __shared__ int      idx_lds[16 * KN];// neighbor index lists
  __shared__ float    qmask[16 * CH];  // staged query mask rows

  const int  lane = threadIdx.x;
  const int  tile = blockIdx.x & 255;          // N/16 = 256 tiles per batch
  const int  b    = blockIdx.x >> 8;
  const int  q0   = tile * 16;
  const int  col  = lane & 15;
  const bool hi   = lane >= 16;

  const float* pcb = pc + (size_t)b * NPTS * 3;
  const float* mb  = mask + (size_t)b * NPTS * CH;

  // ---- A tile: 16 queries x K=4 (x,y,z,0), scaled by -2 ----
  // Layout (ISA 7.12.2, 32-bit A 16x4): lanes 0-15 -> K=0 (v0), K=1 (v1);
  //                                     lanes 16-31 -> K=2 (v0), K=3 (v1).
  const int   q  = q0 + col;
  const float qx = pcb[q * 3 + 0];
  const float qy = pcb[q * 3 + 1];
  const float qz = pcb[q * 3 + 2];
  v2f a;
  a.x = -2.0f * (hi ? qz : qx);
  a.y = hi ? 0.0f : (-2.0f * qy);
  if (!hi) sq_q[col] = qx * qx + qy * qy + qz * qz;
  __syncthreads();

  // Result row for VGPR r at this lane is M = r + (hi ? 8 : 0) -> preload |q|^2.
  float sqrow[8];
#pragma unroll
  for (int r = 0; r < 8; ++r) sqrow[r] = sq_q[r + (hi ? 8 : 0)];

  int cnt   = hi ? KN : 0;   // lanes 16-31 report "done" for the early-exit test
  int first = q0 + lane;     // pad value (self) if somehow nothing found

  for (int m0 = 0; m0 < NPTS; m0 += 16) {
    // ---- B tile: K=4 x 16 candidates (x,y,z,0); lanes = candidate columns ----
    const int   m  = m0 + col;
    const float cx = pcb[m * 3 + 0];
    const float cy = pcb[m * 3 + 1];
    const float cz = pcb[m * 3 + 2];
    const float sqc = cx * cx + cy * cy + cz * cz;
    v2f bv;
    bv.x = hi ? cz : cx;
    bv.y = hi ? 0.0f : cy;

    v8f c = {};
    c = __builtin_amdgcn_wmma_f32_16x16x4_f32(
        /*neg_a=*/false, a, /*neg_b=*/false, bv,
        /*c_mod=*/(short)0, c, /*reuse_a=*/false, /*reuse_b=*/false);

    // Ballot the within-ball flags per result VGPR; bits 0-15 -> row r,
    // bits 16-31 -> row r+8 (C/D layout, ISA 7.12.2).
#pragma unroll
    for (int r = 0; r < 8; ++r) {
      const float d2 = sqrow[r] + sqc + c[r];
      const unsigned bal = (unsigned)__ballot(d2 < R2);
      if (lane == 0) balS[r] = bal;
    }
    __syncthreads();

    // Lane M (< 16) appends this tile's hits for query M in ascending order.
    if (!hi && cnt < KN) {
      unsigned msk = (lane < 8) ? (balS[lane] & 0xFFFFu) : (balS[lane - 8] >> 16);
      while (msk && cnt < KN) {
        const int bit = __builtin_ctz(msk);
        msk &= msk - 1u;
        const int nidx = m0 + bit;
        if (cnt == 0) first = nidx;
        idx_lds[lane * KN + cnt] = nidx;
        ++cnt;
      }
    }
    __syncthreads();
    if (__all(cnt >= KN)) break;   // all 16 queries filled -> stop the scan
  }

  // Pad remaining slots with the first found index (reference semantics).
  if (!hi) {
    for (int s = cnt; s < KN; ++s) idx_lds[lane * KN + s] = first;
  }

  // ---- Phase 2: L1 loss over channels for 16 queries x 16 neighbors ----
  for (int i = lane; i < 16 * CH; i += 32) {
    const int j = i / CH, cc = i % CH;
    qmask[i] = mb[(size_t)(q0 + j) * CH + cc];
  }
  __syncthreads();

  float acc = 0.0f;
  for (int p = lane; p < 16 * KN; p += 32) {
    const int j = p >> 4;                 // query row in tile
    const int s = p & 15;                 // neighbor slot
    const int mi = idx_lds[j * KN + s];
    const float* mm = mb + (size_t)mi * CH;
    const float* qm = qmask + j * CH;
    float ssum = 0.0f;
#pragma unroll
    for (int cc = 0; cc < CH; ++cc) ssum += fabsf(qm[cc] - mm[cc]);
    acc += ssum;
  }

  // Wave32 reduction, then one atomic accumulate of the scaled partial sum.
#pragma unroll
  for (int off = 16; off > 0; off >>= 1) acc += __shfl_down(acc, off);
  if (lane == 0) {
    atomicAdd(out, acc * (1.0f / (float)(KN * BATCH * NPTS)));
  }
}

__global__ void zero_out_kernel(float* out) {
  if (threadIdx.x == 0) out[0] = 0.0f;
}

extern "C" void kernel_launch(void* const* d_in, const int* in_sizes, int n_in,
                              void* d_out, int out_size, void* d_ws, size_t ws_size,
                              hipStream_t stream) {
  (void)in_sizes; (void)n_in; (void)d_ws; (void)ws_size; (void)out_size;
  const float* pc   = (const float*)d_in[0];  // (4, 4096, 3)  f32
  const float* mask = (const float*)d_in[1];  // (4, 4096, 30) f32
  float* out = (float*)d_out;                 // scalar f32

  zero_out_kernel<<<1, 32, 0, stream>>>(out);
  ballq_loss_kernel<<<BATCH * (NPTS / 16), 32, 0, stream>>>(pc, mask, out);
}